// Decoder_73134703116741
// MI455X (gfx1250) — compile-verified
//
#include <hip/hip_runtime.h>
#include <math.h>

// ---------------------------------------------------------------------------
// Decoder step, MI455X (gfx1250, wave32).
// Roofline: enc_feat + enc_out = 2 x 537MB streamed once -> ~46us @ 23.3TB/s.
// GEMMs (<70 MFLOP, shared weights) use V_WMMA_F32_16X16X4_F32.
// Attention Phase A streams enc_feat via TDM (tensor_load_to_lds) double
// buffering into LDS, overlapped with v_tanh_f32 compute.
// ---------------------------------------------------------------------------

#define HH 32
#define NN 64

#if defined(__HIP_DEVICE_COMPILE__) && __has_builtin(__builtin_amdgcn_wmma_f32_16x16x4_f32)
#define HAVE_WMMA_F32 1
#else
#define HAVE_WMMA_F32 0
#endif

#if defined(__HIP_DEVICE_COMPILE__) && __has_builtin(__builtin_amdgcn_tanhf)
#define FAST_TANH(x) __builtin_amdgcn_tanhf(x)
#elif defined(__HIP_DEVICE_COMPILE__) && __has_builtin(__builtin_amdgcn_tanh_f32)
#define FAST_TANH(x) __builtin_amdgcn_tanh_f32(x)
#else
#define FAST_TANH(x) tanhf(x)
#endif

#if defined(__HIP_DEVICE_COMPILE__) &&                         \
    __has_builtin(__builtin_amdgcn_tensor_load_to_lds) &&      \
    __has_builtin(__builtin_amdgcn_s_wait_tensorcnt)
#define HAVE_TDM 1
#else
#define HAVE_TDM 0
#endif

typedef float v2f __attribute__((ext_vector_type(2)));
typedef float v8f __attribute__((ext_vector_type(8)));

__device__ __forceinline__ float sigmoidf_(float x) {
  return 1.0f / (1.0f + __expf(-x));
}

#if HAVE_TDM
typedef unsigned int u32x4 __attribute__((ext_vector_type(4)));
typedef int i32x4 __attribute__((ext_vector_type(4)));
typedef int i32x8 __attribute__((ext_vector_type(8)));

// Issue one TDM 2-D tile load: `rows` rows x 64 f32, row stride 64 f32,
// from gptr into LDS byte-offset lds_off. D# per CDNA5 ISA §8.3/8.4.
__device__ __forceinline__ void tdm_load_rows(unsigned int lds_off,
                                              const float* gptr, int rows) {
  const unsigned long long ga = (unsigned long long)(size_t)gptr;
  u32x4 g0;
  g0.x = 1u;                 // count=1, is_restore=0, gather_mode=0
  g0.y = lds_off;            // lds_addr [63:32]
  g0.z = (unsigned int)ga;   // global_addr [95:64]
  g0.w = (unsigned int)((ga >> 32) & 0x01FFFFFFu) | (2u << 30);  // type=2
  const unsigned int td0 = 64u;       // tensor_dim0
  const unsigned int td1 = 1u << 20;  // tensor_dim1 (large: no OOB clip)
  i32x8 g1;
  g1[0] = (int)(2u << 16);                              // mask=0, data_size=4B
  g1[1] = (int)((td0 & 0xFFFFu) << 16);                 // td0[15:0] -> [63:48]
  g1[2] = (int)((td0 >> 16) | ((td1 & 0xFFFFu) << 16)); // td0 hi | td1 lo
  g1[3] = (int)((td1 >> 16) | (64u << 16));             // td1 hi | tile_dim0=64
  g1[4] = (int)(rows & 0xFFFF);                         // tile_dim1, tile_dim2=0
  g1[5] = 64;                                           // tensor_dim0_stride lo
  g1[6] = 0;                                            // stride0 hi | stride1 lo
  g1[7] = 0;                                            // stride1 hi
  i32x4 z4;
  z4[0] = 0; z4[1] = 0; z4[2] = 0; z4[3] = 0;           // 2-D: groups 2/3 unused
#if __has_include(<hip/amd_detail/amd_gfx1250_TDM.h>)
  // amdgpu-toolchain (therock headers): 6-arg builtin
  i32x8 z8;
  z8[0] = 0; z8[1] = 0; z8[2] = 0; z8[3] = 0;
  z8[4] = 0; z8[5] = 0; z8[6] = 0; z8[7] = 0;
  __builtin_amdgcn_tensor_load_to_lds(g0, g1, z4, z4, z8, 0);
#else
  // ROCm 7.2 (clang-22): 5-arg builtin
  __builtin_amdgcn_tensor_load_to_lds(g0, g1, z4, z4, 0);
#endif
}
#endif  // HAVE_TDM

// ---------------------------------------------------------------------------
// Kernel 1: LSTM step. One wave per batch row, lane j owns hidden unit j.
// ---------------------------------------------------------------------------
__global__ __launch_bounds__(256) void lstm_kernel(
    const float* __restrict__ y, const float* __restrict__ h0,
    const float* __restrict__ c0, const float* __restrict__ W_ih,
    const float* __restrict__ W_hh, const float* __restrict__ b_ih,
    const float* __restrict__ b_hh, float* __restrict__ X,
    float* __restrict__ out_h, float* __restrict__ out_c, int B) {
  const int wave = threadIdx.x >> 5;
  const int lane = threadIdx.x & 31;
  const int b = blockIdx.x * 8 + wave;
  if (b >= B) return;

  const float yb = y[b];
  const float hj = h0[b * HH + lane];
  const float cj = c0[b * HH + lane];

  float g[4];
#pragma unroll
  for (int gi = 0; gi < 4; ++gi) {
    const int row = gi * HH + lane;
    float acc = yb * W_ih[row] + b_ih[row] + b_hh[row];
    const float* wr = W_hh + row * HH;
#pragma unroll
    for (int k = 0; k < HH; ++k) {
      acc += __shfl(hj, k, 32) * wr[k];  // broadcast h across wave32
    }
    g[gi] = acc;
  }
  const float ig = sigmoidf_(g[0]);
  const float fg = sigmoidf_(g[1]);
  const float gg = FAST_TANH(g[2]);
  const float og = sigmoidf_(g[3]);
  const float cn = fg * cj + ig * gg;
  const float hn = og * FAST_TANH(cn);

  out_h[b * HH + lane] = hn;
  out_c[b * HH + lane] = cn;
  X[b * 128 + lane] = hn;  // s_t_hat = [h_new | c_new]
  X[b * 128 + HH + lane] = cn;
}

// ---------------------------------------------------------------------------
// Kernel 2: dec_fea = s_t_hat @ W_proj.T + b_proj   (M=B, K=64, N=64)
// WMMA f32 16x16x4: wave owns 16 rows x 64 cols; 16 K-steps, 4 N-tiles.
// ---------------------------------------------------------------------------
__global__ __launch_bounds__(256) void proj_kernel(
    const float* __restrict__ X, const float* __restrict__ Wp,
    const float* __restrict__ bp, float* __restrict__ dec_fea, int B) {
  const int K = 64, lda = 128;
  const int wave = threadIdx.x >> 5;
  const int lane = threadIdx.x & 31;
  const int half = lane >> 4;
  const int l15 = lane & 15;
  const int b0 = blockIdx.x * 128 + wave * 16;

#if HAVE_WMMA_F32
  v8f acc[4];
#pragma unroll
  for (int nt = 0; nt < 4; ++nt) {
    const float bias = bp[nt * 16 + l15];
#pragma unroll
    for (int i = 0; i < 8; ++i) acc[nt][i] = bias;
  }
  const int m = b0 + l15;
  for (int k = 0; k < K; k += 4) {
    const int kk = k + half * 2;
    const float2 av = *(const float2*)(X + m * lda + kk);
    v2f a;
    a.x = av.x;
    a.y = av.y;
#pragma unroll
    for (int nt = 0; nt < 4; ++nt) {
      const int n = nt * 16 + l15;
      const float2 bv = *(const float2*)(Wp + n * K + kk);
      v2f bb;
      bb.x = bv.x;
      bb.y = bv.y;
      acc[nt] = __builtin_amdgcn_wmma_f32_16x16x4_f32(
          false, a, false, bb, (short)0, acc[nt], false, false);
    }
  }
#pragma unroll
  for (int nt = 0; nt < 4; ++nt) {
    const int col = nt * 16 + l15;
#pragma unroll
    for (int r = 0; r < 8; ++r) {
      dec_fea[(b0 + r + half * 8) * NN + col] = acc[nt][r];
    }
  }
#else
  for (int idx = threadIdx.x; idx < 128 * 64; idx += 256) {
    const int row = blockIdx.x * 128 + (idx >> 6);
    const int c = idx & 63;
    float acc = bp[c];
    for (int k = 0; k < K; ++k) acc += X[row * lda + k] * Wp[c * K + k];
    dec_fea[row * NN + c] = acc;
  }
#endif
}

// ---------------------------------------------------------------------------
// Kernel 3: attention (bandwidth-bound). One block (8 waves) per batch row.
// Phase A: TDM-staged, double-buffered: each wave DMA's its 64-row slice of
//          enc_feat[b] into LDS in 16-row chunks while computing
//          scores[t] = sum_n tanh(ef+df)*v on the previous chunk.
// Phase B: block softmax over T in LDS.
// Phase C: c_t[n] = sum_t attn[t]*enc_out[b,t,n]; lane = feature (coalesced).
// ---------------------------------------------------------------------------
__global__ __launch_bounds__(256) void attn_kernel(
    const float* __restrict__ enc_out, const float* __restrict__ enc_feat,
    const float* __restrict__ dec_fea, const float* __restrict__ vw,
    float* __restrict__ X, int T) {
  __shared__ float s_sc[512];
  __shared__ float s_red[256];
#if HAVE_TDM
  __shared__ float stage[8][2][16 * NN];  // 8 waves x 2 bufs x 4KB = 64KB
#endif
  const int tid = threadIdx.x;
  const int lane = tid & 31;
  const int wave = tid >> 5;
  const int b = blockIdx.x;

  s_sc[tid] = -1e30f;
  s_sc[tid + 256] = -1e30f;
  const float df0 = dec_fea[b * NN + 2 * lane];
  const float df1 = dec_fea[b * NN + 2 * lane + 1];
  const float v0 = vw[2 * lane];
  const float v1 = vw[2 * lane + 1];
  __syncthreads();

  const float* efb = enc_feat + (size_t)b * T * NN;
#if HAVE_TDM
  if (T == 512) {
    // Wave owns rows [wave*64, wave*64+64), staged in 4 chunks of 16 rows.
    const float* wbase = efb + (size_t)(wave * 64) * NN;
    const unsigned int l0 = (unsigned int)(size_t)&stage[wave][0][0];
    const unsigned int l1 = (unsigned int)(size_t)&stage[wave][1][0];
    tdm_load_rows(l0, wbase, 16);
#pragma unroll
    for (int c = 0; c < 4; ++c) {
      if (c + 1 < 4) {
        tdm_load_rows(((c + 1) & 1) ? l1 : l0,
                      wbase + (size_t)(c + 1) * 16 * NN, 16);
        __builtin_amdgcn_s_wait_tensorcnt(1);  // chunk c landed in LDS
      } else {
        __builtin_amdgcn_s_wait_tensorcnt(0);
      }
      asm volatile("" ::: "memory");  // keep LDS reads below the wait
      const float* rbuf = (const float*)&stage[wave][c & 1][0];
#pragma unroll 4
      for (int r = 0; r < 16; ++r) {
        const float2 ef = *(const float2*)(rbuf + r * NN + 2 * lane);
        float s = FAST_TANH(ef.x + df0) * v0 + FAST_TANH(ef.y + df1) * v1;
#pragma unroll
        for (int off = 16; off > 0; off >>= 1) s += __shfl_xor(s, off, 32);
        if (lane == 0) s_sc[wave * 64 + c * 16 + r] = s;
      }
    }
  } else
#endif
  {
    for (int t = wave; t < T; t += 8) {
      const float2* p = (const float2*)(efb + (size_t)t * NN) + lane;
      __builtin_prefetch((const void*)(p + 8 * 32), 0, 1);
      const float2 ef = *p;
      float s = FAST_TANH(ef.x + df0) * v0 + FAST_TANH(ef.y + df1) * v1;
#pragma unroll
      for (int off = 16; off > 0; off >>= 1) s += __shfl_xor(s, off, 32);
      if (lane == 0) s_sc[t] = s;
    }
  }
  __syncthreads();

  // block max
  float m = fmaxf(s_sc[tid], s_sc[tid + 256]);
  s_red[tid] = m;
  __syncthreads();
  for (int s = 128; s > 0; s >>= 1) {
    if (tid < s) s_red[tid] = fmaxf(s_red[tid], s_red[tid + s]);
    __syncthreads();
  }
  const float mx = s_red[0];
  __syncthreads();

  // exp + block sum
  const float e0 = __expf(s_sc[tid] - mx);
  const float e1 = __expf(s_sc[tid + 256] - mx);
  s_red[tid] = e0 + e1;
  __syncthreads();
  for (int s = 128; s > 0; s >>= 1) {
    if (tid < s) s_red[tid] += s_red[tid + s];
    __syncthreads();
  }
  const float inv = 1.0f / s_red[0];
  __syncthreads();
  s_sc[tid] = e0;
  s_sc[tid + 256] = e1;
  __syncthreads();

  // weighted sum over enc_out: 4 t-groups x 64 features
  const int n = tid & 63;
  const int grp = tid >> 6;
  const float* eob = enc_out + (size_t)b * T * NN;
  float acc = 0.f;
  for (int t = grp; t < T; t += 4) {
    const float* p = eob + (size_t)t * NN + n;
    __builtin_prefetch((const void*)(p + 4 * NN), 0, 1);
    acc += s_sc[t] * p[0];
  }
  s_red[tid] = acc;
  __syncthreads();
  if (tid < 64) {
    const float c =
        (s_red[tid] + s_red[tid + 64] + s_red[tid + 128] + s_red[tid + 192]) *
        inv;
    X[b * 128 + NN + tid] = c;  // c_t into out_in[b][64:128]
  }
}

// ---------------------------------------------------------------------------
// Kernel 4: output = (X @ W1.T + b1) @ W2.T + b2.  WMMA GEMM (K=128) fused
// with the final N=64 dot (reduce within 16-lane halves of the C fragment).
// ---------------------------------------------------------------------------
__global__ __launch_bounds__(256) void mlp_kernel(
    const float* __restrict__ X, const float* __restrict__ W1,
    const float* __restrict__ b1, const float* __restrict__ W2,
    const float* __restrict__ b2, float* __restrict__ out, int B) {
  const int K = 128, lda = 128;
  const int wave = threadIdx.x >> 5;
  const int lane = threadIdx.x & 31;
  const int half = lane >> 4;
  const int l15 = lane & 15;
  const int b0 = blockIdx.x * 128 + wave * 16;

#if HAVE_WMMA_F32
  v8f acc[4];
#pragma unroll
  for (int nt = 0; nt < 4; ++nt) {
    const float bias = b1[nt * 16 + l15];
#pragma unroll
    for (int i = 0; i < 8; ++i) acc[nt][i] = bias;
  }
  const int m = b0 + l15;
  for (int k = 0; k < K; k += 4) {
    const int kk = k + half * 2;
    const float2 av = *(const float2*)(X + m * lda + kk);
    v2f a;
    a.x = av.x;
    a.y = av.y;
#pragma unroll
    for (int nt = 0; nt < 4; ++nt) {
      const int n = nt * 16 + l15;
      const float2 bv = *(const float2*)(W1 + n * K + kk);
      v2f bb;
      bb.x = bv.x;
      bb.y = bv.y;
      acc[nt] = __builtin_amdgcn_wmma_f32_16x16x4_f32(
          false, a, false, bb, (short)0, acc[nt], false, false);
    }
  }
  float racc[8] = {0, 0, 0, 0, 0, 0, 0, 0};
#pragma unroll
  for (int nt = 0; nt < 4; ++nt) {
    const float w2v = W2[nt * 16 + l15];
#pragma unroll
    for (int r = 0; r < 8; ++r) racc[r] += acc[nt][r] * w2v;
  }
#pragma unroll
  for (int r = 0; r < 8; ++r) {
#pragma unroll
    for (int off = 1; off < 16; off <<= 1)
      racc[r] += __shfl_xor(racc[r], off, 32);  // within 16-lane half
  }
  if (l15 == 0) {  // lane 0 -> rows b0..b0+7 ; lane 16 -> rows b0+8..b0+15
    const float bb2 = b2[0];
    const int base = b0 + half * 8;
#pragma unroll
    for (int r = 0; r < 8; ++r) out[base + r] = racc[r] + bb2;
  }
#else
  for (int rl = threadIdx.x; rl < 128; rl += 256) {
    const int row = blockIdx.x * 128 + rl;
    float o = b2[0];
    for (int nn = 0; nn < 64; ++nn) {
      float h = b1[nn];
      for (int k = 0; k < K; ++k) h += X[row * lda + k] * W1[nn * K + k];
      o += h * W2[nn];
    }
    out[row] = o;
  }
#endif
}

// ---------------------------------------------------------------------------
extern "C" void kernel_launch(void* const* d_in, const int* in_sizes, int n_in,
                              void* d_out, int out_size, void* d_ws,
                              size_t ws_size, hipStream_t stream) {
  const float* y = (const float*)d_in[0];
  const float* h0 = (const float*)d_in[1];
  const float* c0 = (const float*)d_in[2];
  const float* enc_out = (const float*)d_in[3];
  const float* enc_feat = (const float*)d_in[4];
  const float* W_ih = (const float*)d_in[5];
  const float* W_hh = (const float*)d_in[6];
  const float* b_ih = (const float*)d_in[7];
  const float* b_hh = (const float*)d_in[8];
  const float* W_proj = (const float*)d_in[9];
  const float* b_proj = (const float*)d_in[10];
  const float* v_w = (const float*)d_in[11];
  const float* W1 = (const float*)d_in[12];
  const float* b1 = (const float*)d_in[13];
  const float* W2 = (const float*)d_in[14];
  const float* b2 = (const float*)d_in[15];
  float* out = (float*)d_out;

  const int B = in_sizes[0];             // 4096
  const int T = in_sizes[3] / (B * NN);  // 512

  // workspace: X = out_in [B x 128] (s_t_hat | c_t), dec_fea [B x 64] = 3MB
  float* X = (float*)d_ws;
  float* dec_fea = X + (size_t)B * 128;

  float* out_h = out + B;                   // h_new [1,B,32]
  float* out_c = out + B + (size_t)B * HH;  // c_new [1,B,32]

  lstm_kernel<<<(B + 7) / 8, 256, 0, stream>>>(y, h0, c0, W_ih, W_hh, b_ih,
                                               b_hh, X, out_h, out_c, B);
  proj_kernel<<<B / 128, 256, 0, stream>>>(X, W_proj, b_proj, dec_fea, B);
  attn_kernel<<<B, 256, 0, stream>>>(enc_out, enc_feat, dec_fea, v_w, X, T);
  mlp_kernel<<<B / 128, 256, 0, stream>>>(X, W1, b1, W2, b2, out, B);
}